// HighOrderPairwiseInteraction_1597727834177
// MI455X (gfx1250) — compile-verified
//
#include <hip/hip_runtime.h>
#include <hip/hip_bf16.h>

typedef __attribute__((ext_vector_type(2))) float v2f;
typedef __attribute__((ext_vector_type(4))) float v4f;
typedef __attribute__((ext_vector_type(8))) float v8f;

#define BB 4
#define LL 512
#define RR 512
#define DIN 128
#define DOUT 64

// ---------------------------------------------------------------------------
// Kernel 1: fused projections via V_WMMA_F32_16X16X4_F32.
//   ligb[b*L+l][o] = sum_k lig[b*L+l][k] * W[o][k]       + bias[o]
//   recp[b*R+r][o] = sum_k rec[b*R+r][k] * W[o][128+k]
// One wave (32 lanes) computes one 16x16 tile of a (2048 x 64) projection.
// Waves: [isRec(2)] x [mtile(128)] x [ntile(4)] = 1024 waves.
// ---------------------------------------------------------------------------
__global__ void __launch_bounds__(256)
proj_wmma_kernel(const float* __restrict__ lig,
                 const float* __restrict__ rec,
                 const float* __restrict__ W,     // (64, 256) row-major
                 const float* __restrict__ bias,  // (64)
                 float* __restrict__ ligb,        // (2048, 64)
                 float* __restrict__ recp)        // (2048, 64)
{
    const int wave  = (blockIdx.x * blockDim.x + threadIdx.x) >> 5;
    const int lane  = threadIdx.x & 31;
    const int ntile = wave & 3;          // 4 N-tiles of 16 over DOUT=64
    const int mtile = (wave >> 2) & 127; // 128 M-tiles of 16 over 2048 rows
    const int isRec = wave >> 9;         // 0 = ligand, 1 = receptor

    const float* X     = isRec ? rec : lig;
    float*       P     = isRec ? recp : ligb;
    const int    wcol0 = isRec ? DIN : 0;

    const int lanelo = lane & 15;
    const int khalf  = (lane >> 4) << 1;      // 0 for lanes 0-15, 2 for 16-31

    // A fragment row (M) and B fragment column (N) owned by this lane
    const float* xrow = X + (size_t)(mtile * 16 + lanelo) * DIN;
    const float* wrow = W + (size_t)(ntile * 16 + lanelo) * (2 * DIN) + wcol0;

    v8f c = {0.f, 0.f, 0.f, 0.f, 0.f, 0.f, 0.f, 0.f};

    #pragma unroll 8
    for (int k = 0; k < DIN; k += 4) {
        // ISA 7.12.2: 32-bit A 16x4 — lanes 0-15: VGPR0=K0,VGPR1=K1;
        //                             lanes 16-31: VGPR0=K2,VGPR1=K3.
        v2f a, bf;
        a.x  = xrow[k + khalf + 0];
        a.y  = xrow[k + khalf + 1];
        bf.x = wrow[k + khalf + 0];  // B[k][n] = W[n][k]
        bf.y = wrow[k + khalf + 1];
        c = __builtin_amdgcn_wmma_f32_16x16x4_f32(
                /*neg_a=*/false, a, /*neg_b=*/false, bf,
                /*c_mod=*/(short)0, c, /*reuse_a=*/false, /*reuse_b=*/false);
    }

    // D layout: lanes 0-15 hold rows M=0..7 in VGPR0..7 (N=lane),
    //           lanes 16-31 hold rows M=8..15.
    const int   nn    = ntile * 16 + lanelo;
    const int   mbase = mtile * 16 + ((lane >> 4) << 3);
    const float badd  = isRec ? 0.f : bias[nn];

    #pragma unroll
    for (int i = 0; i < 8; ++i) {
        P[(size_t)(mbase + i) * DOUT + nn] = c[i] + badd;
    }
}

// ---------------------------------------------------------------------------
// Kernel 2: out[b][l][r][o] = ligb[b*L+l][o] + recp[b*R+r][o]   (bias folded)
// One block per (b,l). 256 threads; thread t owns float4 column o4 = t%16
// (invariant across iterations since 256 % 16 == 0) and rows r = t/16 + 16*i.
// 4 KB fully-contiguous store per block per iteration; non-temporal stores
// keep the 268 MB output from thrashing L2 (output > 192 MB L2).
// ---------------------------------------------------------------------------
__global__ void __launch_bounds__(256)
pairwise_add_kernel(const float* __restrict__ ligb,  // (2048, 64)
                    const float* __restrict__ recp,  // (2048, 64)
                    float* __restrict__ outf)        // (2048, 512, 64) floats
{
    const int bl = blockIdx.x;      // b*L + l, 0..2047
    const int b  = bl >> 9;         // L = 512
    const int t  = threadIdx.x;     // 0..255
    const int o4 = t & 15;
    const int r0 = t >> 4;          // 0..15

    const v4f* lig4 = (const v4f*)(ligb + (size_t)bl * DOUT);
    const v4f* rec4 = (const v4f*)(recp + (size_t)b * RR * DOUT);
    v4f* o = (v4f*)(outf + (size_t)bl * RR * DOUT);

    const v4f lv = lig4[o4];

    #pragma unroll 8
    for (int i = 0; i < RR / 16; ++i) {
        const int r = r0 + (i << 4);
        v4f rv = rec4[r * (DOUT / 4) + o4];
        v4f s  = lv + rv;
        __builtin_nontemporal_store(s, &o[(size_t)r * (DOUT / 4) + o4]);
    }
}

extern "C" void kernel_launch(void* const* d_in, const int* in_sizes, int n_in,
                              void* d_out, int out_size, void* d_ws, size_t ws_size,
                              hipStream_t stream) {
    const float* lig  = (const float*)d_in[0];  // (4, 512, 128)
    const float* rec  = (const float*)d_in[1];  // (4, 512, 128)
    const float* W    = (const float*)d_in[2];  // (64, 256)
    const float* bias = (const float*)d_in[3];  // (64)

    float* ligb = (float*)d_ws;                     // (2048, 64) = 512 KB
    float* recp = ligb + (size_t)BB * LL * DOUT;    // (2048, 64) = 512 KB

    // Kernel 1: 1024 waves = 128 blocks x 256 threads (8 waves/block)
    proj_wmma_kernel<<<128, 256, 0, stream>>>(lig, rec, W, bias, ligb, recp);

    // Kernel 2: one block per (b,l) row of the output
    pairwise_add_kernel<<<BB * LL, 256, 0, stream>>>(ligb, recp, (float*)d_out);
}